// c_6_75445395521977
// MI455X (gfx1250) — compile-verified
//
#include <hip/hip_runtime.h>
#include <hip/hip_bf16.h>
#include <math.h>

// ---------------------------------------------------------------------------
//   nodes/depths : [16,5,4,3,224,224] f32  -> 320 images each
//   conv1 3x3 s2 -> [*,3,111,111]; maxpool3 -> [*,3,37,37]; relu
//   conv2 3x3 s2 -> [*,1,18,18];  maxpool3 -> [*,1,6,6];  relu -> 36
//   lin 36->6 per image  => feat_all[640,6]
//   graph msg-passing per batch -> x[16,360]
//   fc1 360->180 relu, fc2 180->60 relu, fc3 60->6  (WMMA f32 16x16x4)
// ---------------------------------------------------------------------------

typedef __attribute__((ext_vector_type(2))) float v2f;
typedef __attribute__((ext_vector_type(8))) float v8f;

#define NIMG 640            // 320 nodes + 320 depths
#define IMW  224
#define WS1_STRIDE 4108     // 3*37*37 = 4107, padded to 16B multiple per image

// ---------------- conv1 + maxpool3 + relu (streaming, LDS-tiled) -----------
// one block per (image g, pooled row ph); blockDim = 128
__global__ __launch_bounds__(128)
void k_conv1(const float* __restrict__ nodes,
             const float* __restrict__ depths,
             const float* __restrict__ c1w, const float* __restrict__ c1b,
             const float* __restrict__ d1w, const float* __restrict__ d1b,
             float* __restrict__ ws1) {
  const int blk = blockIdx.x;
  const int g   = blk / 37;
  const int ph  = blk % 37;
  const bool isN = (g < 320);
  const float* img  = isN ? nodes  + (size_t)g       * 3 * IMW * IMW
                          : depths + (size_t)(g-320) * 3 * IMW * IMW;
  const float* w    = isN ? c1w : d1w;   // [3,3,3,3] = 81
  const float* bias = isN ? c1b : d1b;   // [3]

  __shared__ __align__(16) float sIn[3 * 7 * IMW];
  __shared__ float sW[84];               // 81 weights + 3 bias
  const int tid = threadIdx.x;

  for (int i = tid; i < 84; i += 128)
    sW[i] = (i < 81) ? w[i] : bias[i - 81];

  // stage input rows r0..r0+6 of all 3 channels as float4 (16B-aligned)
  const int r0 = 6 * ph;
  for (int i4 = tid; i4 < 3 * 7 * (IMW / 4); i4 += 128) {
    const int c4 = i4 % (IMW / 4);
    const int t  = i4 / (IMW / 4);
    const int rr = t % 7;
    const int ic = t / 7;
    const float4 v = *(const float4*)(img + (size_t)ic * IMW * IMW
                                          + (size_t)(r0 + rr) * IMW + c4 * 4);
    *(float4*)(&sIn[(ic * 7 + rr) * IMW + c4 * 4]) = v;
  }
  __syncthreads();

  if (tid < 111) {                       // 3 oc * 37 pooled cols
    const int oc = tid / 37;
    const int pw = tid % 37;
    const float* wo = &sW[oc * 27];
    const float  b  = sW[81 + oc];
    float best = -__builtin_inff();
    #pragma unroll
    for (int dy = 0; dy < 3; ++dy) {
      #pragma unroll
      for (int dx = 0; dx < 3; ++dx) {
        const int iy = 2 * dy;           // relative input row
        const int ix = 6 * pw + 2 * dx;  // input col
        float acc = b;
        #pragma unroll
        for (int ic = 0; ic < 3; ++ic)
          #pragma unroll
          for (int ky = 0; ky < 3; ++ky)
            #pragma unroll
            for (int kx = 0; kx < 3; ++kx)
              acc = fmaf(sIn[(ic * 7 + iy + ky) * IMW + ix + kx],
                         wo[ic * 9 + ky * 3 + kx], acc);
        best = fmaxf(best, acc);
      }
    }
    best = fmaxf(best, 0.0f);            // relu after pool
    ws1[(size_t)g * WS1_STRIDE + oc * 37 * 37 + ph * 37 + pw] = best;
  }
}

// ------------- conv2 + maxpool3 + relu + 36->6 linear ----------------------
// one block per image; blockDim = 64
__global__ __launch_bounds__(64)
void k_conv2lin(const float* __restrict__ ws1,
                const float* __restrict__ c2w, const float* __restrict__ c2b,
                const float* __restrict__ d2w, const float* __restrict__ d2b,
                const float* __restrict__ lw,  const float* __restrict__ lb,
                const float* __restrict__ dlw, const float* __restrict__ dlb,
                float* __restrict__ feat_all) {
  const int g = blockIdx.x;
  const bool isN = (g < 320);
  const float* w  = isN ? c2w : d2w;     // [1,3,3,3] = 27
  const float* bp = isN ? c2b : d2b;     // [1]
  const float* LW = isN ? lw  : dlw;     // [6,36]
  const float* LB = isN ? lb  : dlb;     // [6]

  __shared__ __align__(16) float sIn[3 * 37 * 37];
  __shared__ float sP[36];
  __shared__ float sW[28];               // 27 weights + bias
  const int tid = threadIdx.x;

  const float* base = ws1 + (size_t)g * WS1_STRIDE;   // 16B-aligned
  for (int i4 = tid; i4 < 1026; i4 += 64)             // 1026*4 = 4104
    *(float4*)(&sIn[i4 * 4]) = *(const float4*)(base + i4 * 4);
  if (tid < 3) sIn[4104 + tid] = base[4104 + tid];
  if (tid < 27) sW[tid] = w[tid];
  if (tid == 32) sW[27] = bp[0];
  __syncthreads();

  if (tid < 36) {
    const int ph = tid / 6, pw = tid % 6;
    float best = -__builtin_inff();
    #pragma unroll
    for (int dy = 0; dy < 3; ++dy) {
      #pragma unroll
      for (int dx = 0; dx < 3; ++dx) {
        const int r = 3 * ph + dy;       // conv row 0..17
        const int c = 3 * pw + dx;
        float acc = sW[27];
        #pragma unroll
        for (int ic = 0; ic < 3; ++ic)
          #pragma unroll
          for (int ky = 0; ky < 3; ++ky)
            #pragma unroll
            for (int kx = 0; kx < 3; ++kx)
              acc = fmaf(sIn[ic * 37 * 37 + (2 * r + ky) * 37 + (2 * c + kx)],
                         sW[ic * 9 + ky * 3 + kx], acc);
        best = fmaxf(best, acc);
      }
    }
    sP[tid] = fmaxf(best, 0.0f);
  }
  __syncthreads();

  if (tid < 6) {
    float acc = LB[tid];
    #pragma unroll
    for (int j = 0; j < 36; ++j)
      acc = fmaf(sP[j], LW[tid * 36 + j], acc);
    feat_all[(size_t)g * 6 + tid] = acc;
  }
}

// ------------- graph message passing -> x[16,360] --------------------------
// one block per batch element b; blockDim = 384
__global__ __launch_bounds__(384)
void k_buildx(const float* __restrict__ feat_all,
              const float* __restrict__ pos,
              const float* __restrict__ attmap,
              const float* __restrict__ fmp_w, const float* __restrict__ fmp_b,
              const float* __restrict__ lmp_w, const float* __restrict__ lmp_b,
              float* __restrict__ xout) {
  const int b = blockIdx.x;
  __shared__ float sNA[20][12];          // nodes_all
  __shared__ float sMsg[20][12];
  __shared__ float sLast[20][12];
  const int tid = threadIdx.x;

  if (tid < 240) {
    const int j = tid / 12, d = tid % 12;
    sNA[j][d] = (d < 6) ? feat_all[(size_t)(b * 20 + j) * 6 + d]
                        : pos[(size_t)(b * 20 + j) * 6 + (d - 6)];
  }
  __syncthreads();
  if (tid < 240) {
    const int j = tid / 12, d = tid % 12;
    float m = fmp_b[d], l = lmp_b[d];
    #pragma unroll
    for (int k = 0; k < 12; ++k) {
      const float v = sNA[j][k];
      m = fmaf(v, fmp_w[d * 12 + k], m);
      l = fmaf(v, lmp_w[d * 12 + k], l);
    }
    sMsg[j][d] = m; sLast[j][d] = l;
  }
  __syncthreads();
  if (tid < 360) {
    const int jn = tid / 18, c = tid % 18;   // node index, channel in 18-chunk
    const int f = jn / 4, n = jn % 4;
    float v;
    if (c < 12) {
      float acc = 0.0f;
      #pragma unroll
      for (int m = 0; m < 4; ++m)
        acc = fmaf(attmap[((size_t)(b * 5 + f) * 4 + m) * 4 + n], sMsg[f * 4 + m][c], acc);
      if (f > 0) acc += sLast[(f - 1) * 4 + n][c];
      v = acc;
    } else {
      v = feat_all[(size_t)(320 + b * 20 + jn) * 6 + (c - 12)];  // dfeat
    }
    xout[(size_t)b * 360 + tid] = v;
  }
}

// ------------- FC head with V_WMMA_F32_16X16X4_F32 -------------------------
// one block, 12 waves (384 threads). M=16 = batch.
// A frag (16x4): lanes 0-15 hold K=kk,kk+1 for row=l16; lanes 16-31 hold K=kk+2,kk+3.
// C/D:  VGPR r: lanes 0-15 -> M=r, N=lane; lanes 16-31 -> M=r+8, N=lane-16.
// Tail columns handled by index clamp + multiplicative zero mask (no EXEC
// divergence -> WMMA always executes with full EXEC).
__device__ __forceinline__
v8f fc_tile(const float* __restrict__ sA, const float* __restrict__ W,
            int K, int n, int nValid, int l16, int hs) {
  const bool  nv  = (n < nValid);
  const float msk = nv ? 1.0f : 0.0f;
  const int   ns  = nv ? n : 0;
  v8f c = {};
  for (int kk = 0; kk < K; kk += 4) {
    const int kb = kk + hs * 2;
    v2f a  = *(const v2f*)(sA + l16 * K + kb);             // ds_load_b64
    v2f bf = *(const v2f*)(W + (size_t)ns * K + kb);       // global_load_b64
    bf *= msk;                                             // zero invalid cols
    c = __builtin_amdgcn_wmma_f32_16x16x4_f32(false, a, false, bf,
                                              (short)0, c, false, false);
  }
  return c;
}

__global__ __launch_bounds__(384)
void k_fc(const float* __restrict__ xin,
          const float* __restrict__ fc1w, const float* __restrict__ fc1b,
          const float* __restrict__ fc2w, const float* __restrict__ fc2b,
          const float* __restrict__ fc3w, const float* __restrict__ fc3b,
          float* __restrict__ out) {
  __shared__ __align__(16) float sX [16 * 360];
  __shared__ __align__(16) float sH1[16 * 180];
  __shared__ __align__(16) float sH2[16 * 60];
  const int tid  = threadIdx.x;
  const int wave = tid >> 5;
  const int lane = tid & 31;
  const int hs   = lane >> 4;            // 0: K lo pair, 1: K hi pair
  const int l16  = lane & 15;

  for (int i4 = tid; i4 < (16 * 360) / 4; i4 += 384)
    *(float4*)(&sX[i4 * 4]) = *(const float4*)(xin + i4 * 4);
  __syncthreads();

  // ---- fc1: [16,360] x [360,180]^T ; 12 tiles of 16 cols (last tile 4 valid)
  {
    const int n = wave * 16 + l16;
    v8f c = fc_tile(sX, fc1w, 360, n, 180, l16, hs);
    if (n < 180) {
      const float bias = fc1b[n];
      #pragma unroll
      for (int r = 0; r < 8; ++r) {
        const int m = r + hs * 8;
        const float v = c[r] + bias;
        sH1[m * 180 + n] = v > 0.0f ? v : 0.0f;
      }
    }
  }
  __syncthreads();

  // ---- fc2: [16,180] x [180,60]^T ; waves 0..3 (last tile 12 valid)
  if (wave < 4) {
    const int n = wave * 16 + l16;
    v8f c = fc_tile(sH1, fc2w, 180, n, 60, l16, hs);
    if (n < 60) {
      const float bias = fc2b[n];
      #pragma unroll
      for (int r = 0; r < 8; ++r) {
        const int m = r + hs * 8;
        const float v = c[r] + bias;
        sH2[m * 60 + n] = v > 0.0f ? v : 0.0f;
      }
    }
  }
  __syncthreads();

  // ---- fc3: [16,60] x [60,6]^T ; wave 0 only
  if (wave == 0) {
    const int n = l16;
    v8f c = fc_tile(sH2, fc3w, 60, n, 6, l16, hs);
    if (n < 6) {
      const float bias = fc3b[n];
      #pragma unroll
      for (int r = 0; r < 8; ++r) {
        const int m = r + hs * 8;
        out[m * 6 + n] = c[r] + bias;
      }
    }
  }
}

// ---------------------------------------------------------------------------
extern "C" void kernel_launch(void* const* d_in, const int* in_sizes, int n_in,
                              void* d_out, int out_size, void* d_ws, size_t ws_size,
                              hipStream_t stream) {
  const float* nodes   = (const float*)d_in[0];
  const float* pos     = (const float*)d_in[1];
  const float* attmap  = (const float*)d_in[2];
  const float* depths  = (const float*)d_in[3];
  const float* conv1_w = (const float*)d_in[4];
  const float* conv1_b = (const float*)d_in[5];
  const float* conv2_w = (const float*)d_in[6];
  const float* conv2_b = (const float*)d_in[7];
  const float* lin_w   = (const float*)d_in[8];
  const float* lin_b   = (const float*)d_in[9];
  const float* dconv1_w= (const float*)d_in[10];
  const float* dconv1_b= (const float*)d_in[11];
  const float* dconv2_w= (const float*)d_in[12];
  const float* dconv2_b= (const float*)d_in[13];
  const float* dlin_w  = (const float*)d_in[14];
  const float* dlin_b  = (const float*)d_in[15];
  const float* fmp_w   = (const float*)d_in[16];
  const float* fmp_b   = (const float*)d_in[17];
  const float* lmp_w   = (const float*)d_in[18];
  const float* lmp_b   = (const float*)d_in[19];
  const float* fc1_w   = (const float*)d_in[20];
  const float* fc1_b   = (const float*)d_in[21];
  const float* fc2_w   = (const float*)d_in[22];
  const float* fc2_b   = (const float*)d_in[23];
  const float* fc3_w   = (const float*)d_in[24];
  const float* fc3_b   = (const float*)d_in[25];

  float* ws       = (float*)d_ws;
  float* ws1      = ws;                                   // [640, WS1_STRIDE]
  float* feat_all = ws1 + (size_t)NIMG * WS1_STRIDE;      // [640,6]
  float* xbuf     = feat_all + (size_t)NIMG * 6;          // [16,360]

  k_conv1<<<dim3(NIMG * 37), dim3(128), 0, stream>>>(
      nodes, depths, conv1_w, conv1_b, dconv1_w, dconv1_b, ws1);

  k_conv2lin<<<dim3(NIMG), dim3(64), 0, stream>>>(
      ws1, conv2_w, conv2_b, dconv2_w, dconv2_b,
      lin_w, lin_b, dlin_w, dlin_b, feat_all);

  k_buildx<<<dim3(16), dim3(384), 0, stream>>>(
      feat_all, pos, attmap, fmp_w, fmp_b, lmp_w, lmp_b, xbuf);

  k_fc<<<dim3(1), dim3(384), 0, stream>>>(
      xbuf, fc1_w, fc1_b, fc2_w, fc2_b, fc3_w, fc3_b, (float*)d_out);
}